// SSConnectionLearning_29910152249947
// MI455X (gfx1250) — compile-verified
//
#include <hip/hip_runtime.h>
#include <hip/hip_bf16.h>
#include <math.h>

#define H_ 4
#define N_ 512
#define D_ 64
#define E_ (N_ * N_)
#define EPSG 1e-10f

typedef __attribute__((ext_vector_type(2))) float v2f;
typedef __attribute__((ext_vector_type(8))) float v8f;

// ---------------------------------------------------------------------------
// Kernel 1: per-node partial GEMMs via V_WMMA_F32_16X16X4_F32 (full fp32).
//   PA [h,n,d] = feats[h,n,:] @ W_out[0:64 , d]
//   PBb[h,n,d] = feats[h,n,:] @ W_out[64:128, d] + b_out[d]
// One wave32 per 16x16 output tile; 16 WMMA steps cover K=64.
// Tiles: H(4) * half(2) * Mtiles(32) * Ntiles(4) = 1024 -> 256 blocks x 4 waves.
// ---------------------------------------------------------------------------
__global__ __launch_bounds__(128) void node_partials_wmma(
    const float* __restrict__ feats, const float* __restrict__ W_out,
    const float* __restrict__ b_out, float* __restrict__ PA,
    float* __restrict__ PBb)
{
    const int wave = threadIdx.x >> 5;
    const int lane = threadIdx.x & 31;
    int tile = blockIdx.x * 4 + wave;        // 0..1023
    const int ntile = tile & 3;  tile >>= 2; // N tile (0..3)
    const int mtile = tile & 31; tile >>= 5; // M tile (0..31)
    const int half  = tile & 1;  tile >>= 1; // 0 = sender weights, 1 = receiver
    const int h     = tile;                  // head

    const int l15    = lane & 15;
    const int hiHalf = lane >> 4;            // 0: K,K+1  1: K+2,K+3
    const int m  = mtile * 16 + l15;         // A row for this lane
    const int n0 = ntile * 16 + l15;         // B/D column for this lane

    const float* fbase = feats + (size_t)(h * N_ + m) * D_;
    const float* wbase = W_out + (size_t)(half * D_) * D_ + n0;

    v8f c = {};
#pragma unroll
    for (int k0 = 0; k0 < D_; k0 += 4) {
        const int ka = k0 + hiHalf * 2;
        v2f a, b;
        a.x = fbase[ka];
        a.y = fbase[ka + 1];
        b.x = wbase[(size_t)ka * D_];
        b.y = wbase[(size_t)(ka + 1) * D_];
        // 8 args: (neg_a, A, neg_b, B, c_mod, C, reuse_a, reuse_b)
        c = __builtin_amdgcn_wmma_f32_16x16x4_f32(false, a, false, b,
                                                  (short)0, c, false, false);
    }

    const float bias = half ? b_out[n0] : 0.0f;
    float* dst = half ? PBb : PA;
#pragma unroll
    for (int i = 0; i < 8; ++i) {
        const int row = mtile * 16 + i + hiHalf * 8;  // C/D layout: VGPR i -> M=i / M=i+8
        dst[(size_t)(h * N_ + row) * D_ + n0] = c[i] + bias;
    }
}

// ---------------------------------------------------------------------------
// Deterministic hash -> Gumbel noise (stand-in for jax.random with key 42).
// ---------------------------------------------------------------------------
__device__ __forceinline__ float gumbel_hash(unsigned x)
{
    x ^= 42u * 0x9E3779B9u;
    x *= 0x85EBCA6Bu; x ^= x >> 13;
    x *= 0xC2B2AE35u; x ^= x >> 16;
    const float U = (float)(x >> 8) * (1.0f / 16777216.0f); // [0,1)
    return -__logf(-__logf(U + EPSG) + EPSG);
}

// ---------------------------------------------------------------------------
// Kernel 2: one block per (head, receiver). 512 senders per block.
//   hvec = relu(PA[snd] + PBb[rec]) ; logits = hvec @ W_cat + b_cat
//   prob = sigmoid(l1 - l0) ; hard = argmax(logits + gumbel) == 1
// ---------------------------------------------------------------------------
__global__ __launch_bounds__(256) void edge_kernel(
    const float* __restrict__ PA, const float* __restrict__ PBb,
    const float* __restrict__ W_cat, const float* __restrict__ b_cat,
    float* __restrict__ out_cm, float* __restrict__ out_prob,
    float* __restrict__ out_sl)
{
    __shared__ float pbr[D_], wc0[D_], wc1[D_], bc[2];
    const int h = blockIdx.x >> 9;
    const int r = blockIdx.x & (N_ - 1);
    const int t = threadIdx.x;

    if (t < D_) {
        pbr[t] = PBb[(size_t)(h * N_ + r) * D_ + t];
        wc0[t] = W_cat[t * 2 + 0];
        wc1[t] = W_cat[t * 2 + 1];
    }
    if (t < 2) bc[t] = b_cat[t];
    __syncthreads();

    for (int s = t; s < N_; s += 256) {
        const float4* pa = (const float4*)(PA + (size_t)(h * N_ + s) * D_);
        float l0 = bc[0], l1 = bc[1];
#pragma unroll
        for (int q = 0; q < D_ / 4; ++q) {
            const float4 v = pa[q];
            const int d = q * 4;
            float h0 = v.x + pbr[d + 0]; h0 = h0 > 0.f ? h0 : 0.f;
            float h1 = v.y + pbr[d + 1]; h1 = h1 > 0.f ? h1 : 0.f;
            float h2 = v.z + pbr[d + 2]; h2 = h2 > 0.f ? h2 : 0.f;
            float h3 = v.w + pbr[d + 3]; h3 = h3 > 0.f ? h3 : 0.f;
            l0 = fmaf(h0, wc0[d + 0], l0); l1 = fmaf(h0, wc1[d + 0], l1);
            l0 = fmaf(h1, wc0[d + 1], l0); l1 = fmaf(h1, wc1[d + 1], l1);
            l0 = fmaf(h2, wc0[d + 2], l0); l1 = fmaf(h2, wc1[d + 2], l1);
            l0 = fmaf(h3, wc0[d + 3], l0); l1 = fmaf(h3, wc1[d + 3], l1);
        }
        const float prob = 1.0f / (1.0f + __expf(l0 - l1)); // softmax(..)[1]

        const size_t idx = (size_t)h * E_ + (size_t)r * N_ + s;
        const float g0 = gumbel_hash((unsigned)(idx * 2 + 0));
        const float g1 = gumbel_hash((unsigned)(idx * 2 + 1));
        // temperature > 0 does not change the argmax; ties -> class 0
        const float hard = ((l1 + g1) > (l0 + g0)) ? 1.0f : 0.0f;

        out_cm[idx]   = hard;   // y_hard[...,1] (straight-through fwd value)
        out_prob[idx] = prob;
        out_sl[idx]   = 0.0f;   // zero-init; kernel 3 sets the top-K ones
    }
}

// ---------------------------------------------------------------------------
// Kernel 3: per-head exact top-K via binary search on positive-float bits.
// One block per head; probs in (0,1) so bit patterns are order-preserving.
// ---------------------------------------------------------------------------
__global__ __launch_bounds__(256) void topk_kernel(
    const float* __restrict__ out_prob, float* __restrict__ out_sl,
    const int* __restrict__ sparse_k)
{
    const int h = blockIdx.x;
    const int t = threadIdx.x;
    const unsigned K = (unsigned)(*sparse_k);
    const float* probs = out_prob + (size_t)h * E_;
    float* sl = out_sl + (size_t)h * E_;

    __shared__ unsigned red[256];
    __shared__ unsigned s_lo, s_hi, s_c1;
    __shared__ int s_cnt;

    if (t == 0) { s_lo = 0u; s_hi = 0x3F800001u; }
    __syncthreads();

    // v = max u such that count(bits >= u) >= K   (v == bits of K-th largest)
    for (;;) {
        const unsigned lo = s_lo, hi = s_hi;
        if (lo + 1 >= hi) break;
        const unsigned mid = (lo + hi) >> 1;
        unsigned cnt = 0;
        for (int i = t; i < E_; i += 256)
            cnt += (__float_as_uint(probs[i]) >= mid) ? 1u : 0u;
        red[t] = cnt;
        __syncthreads();
        for (int st = 128; st > 0; st >>= 1) {
            if (t < st) red[t] += red[t + st];
            __syncthreads();
        }
        if (t == 0) { if (red[0] >= K) s_lo = mid; else s_hi = mid; }
        __syncthreads();
    }
    const unsigned v = s_lo;

    // c1 = strictly-greater count
    unsigned cnt = 0;
    for (int i = t; i < E_; i += 256)
        cnt += (__float_as_uint(probs[i]) > v) ? 1u : 0u;
    red[t] = cnt;
    __syncthreads();
    for (int st = 128; st > 0; st >>= 1) {
        if (t < st) red[t] += red[t + st];
        __syncthreads();
    }
    if (t == 0) { s_c1 = red[0]; s_cnt = 0; }
    __syncthreads();

    const int need = (int)K - (int)s_c1; // ties at v to admit
    for (int i = t; i < E_; i += 256) {
        const unsigned b = __float_as_uint(probs[i]);
        if (b > v) {
            sl[i] = 1.0f;
        } else if (b == v) {
            const int pos = atomicAdd(&s_cnt, 1);
            if (pos < need) sl[i] = 1.0f;
        }
    }
}

// ---------------------------------------------------------------------------
extern "C" void kernel_launch(void* const* d_in, const int* in_sizes, int n_in,
                              void* d_out, int out_size, void* d_ws,
                              size_t ws_size, hipStream_t stream)
{
    const float* feats    = (const float*)d_in[0]; // [H,N,D]
    const float* W_out    = (const float*)d_in[1]; // [2D,D]
    const float* b_out    = (const float*)d_in[2]; // [D]
    const float* W_cat    = (const float*)d_in[3]; // [D,2]
    const float* b_cat    = (const float*)d_in[4]; // [2]
    const int*   sparse_k = (const int*)d_in[5];   // scalar

    float* PA  = (float*)d_ws;                     // [H,N,D]
    float* PBb = PA + (size_t)H_ * N_ * D_;        // [H,N,D] (+bias)

    float* out_cm   = (float*)d_out;               // [H,N,N]
    float* out_prob = out_cm + (size_t)H_ * E_;    // [H,E]
    float* out_sl   = out_prob + (size_t)H_ * E_;  // [H,E]

    node_partials_wmma<<<256, 128, 0, stream>>>(feats, W_out, b_out, PA, PBb);
    edge_kernel<<<H_ * N_, 256, 0, stream>>>(PA, PBb, W_cat, b_cat,
                                             out_cm, out_prob, out_sl);
    topk_kernel<<<H_, 256, 0, stream>>>(out_prob, out_sl, sparse_k);
}